// FNO3d_25426206392438
// MI455X (gfx1250) — compile-verified
//
#include <hip/hip_runtime.h>
#include <hip/hip_bf16.h>
#include <math.h>

// FNO3d on MI455X (gfx1250): all FFT stages expressed as partial-DFT GEMMs
// on V_WMMA_F32_16X16X4_F32. Only the 4 corner mode blocks are computed
// (reference keeps .real of complex*realweight => only Re(x_ft) at 64x64x32
// modes matters). All stage constants are template parameters so the K-loop
// has zero address multiplies: pointers advance by compile-time strides.
//
// Pipeline (all f32, Re/Im stacked into real GEMMs):
//   S1: z-DFT    : [65536 x 256] x [256 x 64]           -> A1 [65536][64]
//   S2: y-DFT    : batch x=256: T2[128x512] x B[512x32] -> O2 [256][128][32]
//   S3: x-DFT(Re): batch ky=64: T3[64x512] x B[512x32]  -> C3 [64][64][32]
//   G : corner weight multiply (w1..w4)
//   I1: x-iDFT   : batch ky=64: E1[512x64] x G[64x32]   -> T1 [64][512][32]
//   I2: y-iDFT   : batch x=256: E2[512x128] x B[128x32] -> T2F [65536][64]
//   I3: z-irfft  : [65536 x 64] x B4[64 x 256] fused W0/gelu/W4 epilogue.
// Intermediates (<17MB each) stay resident in the 192MB L2.

typedef __attribute__((ext_vector_type(2))) float v2f;
typedef __attribute__((ext_vector_type(8))) float v8f;

// ---- workspace layout (float offsets) ----
#define OFF_F1   0L          // [256][64]
#define OFF_T2   16384L      // [128][512]
#define OFF_T3   81920L      // [64][512]
#define OFF_E1   114688L     // [512][64]
#define OFF_E2   147456L     // [512][128]
#define OFF_B4   212992L     // [64][256]
#define OFF_A1   229376L     // [65536][64]
#define OFF_O2   4423680L    // [256][128][32]
#define OFF_C3   5472256L    // [64][64][32]
#define OFF_G    5603328L    // [64][64][32]
#define OFF_T1   5734400L    // [64][512][32]
#define OFF_T2F  6782976L    // [65536][64]
// total = 10977280 floats = 41.9 MiB

#define PI2_OVER_256 0.0245436926061703f

// ---------------- twiddle/basis generation ----------------
__global__ void fno_init_bases(float* ws) {
  int id = blockIdx.x * blockDim.x + threadIdx.x;
  const float step = PI2_OVER_256;
  if (id < 16384) {                       // F1 [k=256][n=64]: Re=cos, Im=-sin (e^{-i})
    int k = id >> 6, n = id & 63, f = n & 31;
    float th = (float)((f * k) & 255) * step;
    ws[OFF_F1 + id] = (n < 32) ? cosf(th) : -sinf(th);
  } else if (id < 81920) {                // T2 [j=128][k=512]: [[c,s],[-s,c]]
    int l = id - 16384;
    int j = l >> 9, k = l & 511, jj = j & 63;
    int f = (jj < 32) ? jj : (192 + jj);  // Ky
    float th = (float)((f * (k & 255)) & 255) * step;
    float c = cosf(th), s = sinf(th);
    ws[OFF_T2 + l] = (j < 64) ? ((k < 256) ? c : s) : ((k < 256) ? -s : c);
  } else if (id < 114688) {               // T3 [i=64][k=512]: [c | s] (Re only)
    int l = id - 81920;
    int i = l >> 9, k = l & 511;
    int f = (i < 32) ? i : (192 + i);     // Kx
    float th = (float)((f * (k & 255)) & 255) * step;
    ws[OFF_T3 + l] = (k < 256) ? cosf(th) : sinf(th);
  } else if (id < 147456) {               // E1 [q=512][i=64]: rows<256 cos, >=256 sin (e^{+i})
    int l = id - 114688;
    int q = l >> 6, i = l & 63;
    int f = (i < 32) ? i : (192 + i);
    float th = (float)((f * (q & 255)) & 255) * step;
    ws[OFF_E1 + l] = (q < 256) ? cosf(th) : sinf(th);
  } else if (id < 212992) {               // E2 [q=512][k=128]: [[c,-s],[s,c]]
    int l = id - 147456;
    int q = l >> 7, k = l & 127, jj = k & 63;
    int f = (jj < 32) ? jj : (192 + jj);
    float th = (float)((f * (q & 255)) & 255) * step;
    float c = cosf(th), s = sinf(th);
    ws[OFF_E2 + l] = (q < 256) ? ((k < 64) ? c : -s) : ((k < 64) ? s : c);
  } else if (id < 229376) {               // B4 [k=64][z=256]: irfft row (g*cos | -g*sin), g folds 1/256^3
    int l = id - 212992;
    int k = l >> 8, z = l & 255, kz = k & 31;
    float g = ((kz == 0) ? 1.0f : 2.0f) * (1.0f / 16777216.0f);
    float th = (float)((kz * z) & 255) * step;
    ws[OFF_B4 + l] = (k < 32) ? g * cosf(th) : -g * sinf(th);
  }
}

// ---------------- WMMA GEMM core ----------------
// A: row-major [M x K], element A[aOff + row*LDA + k].
// B: piecewise-affine over K: off(k) = bOff + (k % KSEG)*KS_LO + (k / KSEG)*KS_HI + col
//    (K == NSEG*KSEG, NSEG <= 2 in all stages). Pointers advance by
//    compile-time strides; no multiplies in the loop.
// Computes a 16 x (16*NACC) strip, reusing the A fragment NACC times.
template<int K, int KSEG, int KS_LO, int KS_HI, int LDA, int NACC>
__device__ __forceinline__ void fno_wmma_core(
    const float* __restrict__ A, const float* __restrict__ B,
    long aOff, long bOff, int row0, int col0, v8f (&acc)[NACC])
{
  int lane = threadIdx.x & 31;
  int half = lane >> 4;                 // 0: K pair (0,1); 1: K pair (2,3)
  int l15  = lane & 15;
  const float* ap = A + aOff + (long)(row0 + l15) * LDA + (half << 1);
  const float* bp = B + bOff + (long)(half << 1) * KS_LO + (col0 + l15);
  constexpr int NSEG = K / KSEG;
#pragma unroll
  for (int seg = 0; seg < NSEG; ++seg) {
#pragma unroll 4
    for (int k0 = 0; k0 < KSEG; k0 += 4) {
      v2f a = *reinterpret_cast<const v2f*>(ap);
#pragma unroll
      for (int t = 0; t < NACC; ++t) {
        v2f b;
        b.x = bp[t * 16];
        b.y = bp[t * 16 + KS_LO];
        acc[t] = __builtin_amdgcn_wmma_f32_16x16x4_f32(
            false, a, false, b, (short)0, acc[t], false, false);
      }
      ap += 4;
      bp += 4 * KS_LO;
    }
    if (seg + 1 < NSEG) bp += KS_HI - (long)KSEG * KS_LO;  // jump to next segment base
  }
}

// ---------------- fully-specialized batched GEMM kernel ----------------
// One wave per 16x(16*NACC) strip. C map:
//   flat = CB*b + (row&RMASK)*RS_LO + (row>>RSHIFT)*RS_HI + col
template<int K, int KSEG, int KS_LO, int KS_HI, int LDA, int NACC,
         int MT, int NTW, int BATCH, long AB, long BB, long CB,
         int RMASK, int RSHIFT, int RS_LO, int RS_HI>
__global__ void fno_gemm(const float* __restrict__ A,
                         const float* __restrict__ B,
                         float* __restrict__ C)
{
  int wave = blockIdx.x * (blockDim.x >> 5) + (threadIdx.x >> 5);
  constexpr int TPB = MT * NTW;
  int b = wave / TPB;
  if (b >= BATCH) return;
  int t = wave - b * TPB;
  int mt = t / NTW, ntw = t - mt * NTW;

  v8f acc[NACC];
#pragma unroll
  for (int i = 0; i < NACC; ++i) acc[i] = 0.0f;

  fno_wmma_core<K, KSEG, KS_LO, KS_HI, LDA, NACC>(
      A, B, AB * b, BB * b, mt * 16, ntw * 16 * NACC, acc);

  int lane = threadIdx.x & 31;
  int l15 = lane & 15, half = lane >> 4;
  long cb = CB * b;
#pragma unroll
  for (int ta = 0; ta < NACC; ++ta) {
    int col = ntw * 16 * NACC + ta * 16 + l15;
#pragma unroll
    for (int j = 0; j < 8; ++j) {
      int row = mt * 16 + j + (half << 3);
      long off = cb + (long)(row & RMASK) * RS_LO +
                 (long)((unsigned)row >> RSHIFT) * RS_HI + col;
      C[off] = acc[ta][j];
    }
  }
}

// ---------------- corner-mode weight multiply ----------------
__global__ void fno_mode_mul(const float* __restrict__ ws_in /* C3 */,
                             float* __restrict__ ws_out /* G */,
                             const float* __restrict__ w1,
                             const float* __restrict__ w2,
                             const float* __restrict__ w3,
                             const float* __restrict__ w4) {
  int id = blockIdx.x * blockDim.x + threadIdx.x;  // 131072
  int i  = id >> 11;          // kx index 0..63
  int j  = (id >> 5) & 63;    // ky index 0..63
  int kz = id & 31;
  int sel = ((j >> 5) << 1) | (i >> 5);
  const float* w = (sel == 0) ? w1 : (sel == 1) ? w2 : (sel == 2) ? w3 : w4;
  int widx = (((i & 31) * 32) + (j & 31)) * 32 + kz;
  ws_out[id] = ws_in[id] * w[widx];
}

// ---------------- final GEMM + fused epilogue ----------------
// x1 = T2F[65536x64] * B4[64x256]; v = x1 + W0*pot + b0; g = gelu_tanh(v);
// out[xy][z][c] = g*W4[c] + b4[c]
__global__ void fno_gemm_final(
    const float* __restrict__ A /* T2F */, const float* __restrict__ B4,
    const float* __restrict__ pot,
    const float* __restrict__ W0, const float* __restrict__ b0,
    const float* __restrict__ W4, const float* __restrict__ b4,
    float* __restrict__ out)
{
  int wave = blockIdx.x * (blockDim.x >> 5) + (threadIdx.x >> 5);
  // M=65536 (MT=4096), N=256 with NACC=4 -> NTW=4
  int mt = wave >> 2, ntw = wave & 3;
  if (mt >= 4096) return;

  v8f acc[4];
#pragma unroll
  for (int i = 0; i < 4; ++i) acc[i] = 0.0f;

  fno_wmma_core<64, 64, 256, 0, 64, 4>(A, B4, 0L, 0L, mt * 16, ntw * 64, acc);

  float w0 = W0[0], bb0 = b0[0];
  float w40 = W4[0], w41 = W4[1], w42 = W4[2];
  float c40 = b4[0], c41 = b4[1], c42 = b4[2];

  int lane = threadIdx.x & 31;
  int l15 = lane & 15, half = lane >> 4;
#pragma unroll
  for (int ta = 0; ta < 4; ++ta) {
    int z = ntw * 64 + ta * 16 + l15;
#pragma unroll
    for (int j = 0; j < 8; ++j) {
      int xy = mt * 16 + j + (half << 3);
      long idx = (long)xy * 256 + z;
      float v = acc[ta][j] + w0 * pot[idx] + bb0;
      float u = 0.7978845608f * (v + 0.044715f * v * v * v);
      float g = 0.5f * v * (1.0f + tanhf(u));      // jax.nn.gelu (tanh approx)
      long o = idx * 3;
      out[o + 0] = g * w40 + c40;
      out[o + 1] = g * w41 + c41;
      out[o + 2] = g * w42 + c42;
    }
  }
}

extern "C" void kernel_launch(void* const* d_in, const int* in_sizes, int n_in,
                              void* d_out, int out_size, void* d_ws, size_t ws_size,
                              hipStream_t stream) {
  (void)in_sizes; (void)n_in; (void)out_size; (void)ws_size;
  const float* pot = (const float*)d_in[0];
  const float* w1  = (const float*)d_in[1];
  const float* w2  = (const float*)d_in[2];
  const float* w3  = (const float*)d_in[3];
  const float* w4  = (const float*)d_in[4];
  const float* W0  = (const float*)d_in[5];
  const float* b0  = (const float*)d_in[6];
  const float* W4  = (const float*)d_in[7];
  const float* b4  = (const float*)d_in[8];
  float* out = (float*)d_out;
  float* ws  = (float*)d_ws;

  const int BLK = 128;           // 4 waves/block
  constexpr int BM = 0x7fffffff; // "no split" row mask

  // twiddle bases
  fno_init_bases<<<896, 256, 0, stream>>>(ws);

  // S1: z-DFT. M=65536(MT=4096), N=64(NACC=4,NTW=1), K=256, batch=1. waves=4096
  fno_gemm<256, 256, 64, 0, 256, 4, 4096, 1, 1, 0L, 0L, 0L, BM, 30, 64, 0>
      <<<1024, BLK, 0, stream>>>(pot, ws + OFF_F1, ws + OFF_A1);

  // S2: y-DFT. M=128(MT=8), N=32(NACC=2), K=512(2 segs of 256), batch=256. waves=2048
  fno_gemm<512, 256, 64, 32, 512, 2, 8, 1, 256, 0L, 16384L, 4096L, BM, 30, 32, 0>
      <<<512, BLK, 0, stream>>>(ws + OFF_T2, ws + OFF_A1, ws + OFF_O2);

  // S3: x-DFT(Re). M=64(MT=4), N=32(NACC=2), K=512(2 segs), batch=64. waves=256
  fno_gemm<512, 256, 4096, 2048, 512, 2, 4, 1, 64, 0L, 32L, 32L, BM, 30, 2048, 0>
      <<<64, BLK, 0, stream>>>(ws + OFF_T3, ws + OFF_O2, ws + OFF_C3);

  // G: corner weight multiply (131072 elements)
  fno_mode_mul<<<512, 256, 0, stream>>>(ws + OFF_C3, ws + OFF_G, w1, w2, w3, w4);

  // I1: x-iDFT. M=512(MT=32), N=32(NACC=2), K=64, batch=64. waves=2048
  fno_gemm<64, 64, 2048, 0, 64, 2, 32, 1, 64, 0L, 32L, 16384L, BM, 30, 32, 0>
      <<<512, BLK, 0, stream>>>(ws + OFF_E1, ws + OFF_G, ws + OFF_T1);

  // I2: y-iDFT. M=512(MT=32), N=32(NACC=2), K=128(2 segs of 64), batch=256. waves=8192
  // C map: flat = b*16384 + (row&255)*64 + (row>>8)*32 + col
  fno_gemm<128, 64, 16384, 8192, 128, 2, 32, 1, 256, 0L, 32L, 16384L, 255, 8, 64, 32>
      <<<2048, BLK, 0, stream>>>(ws + OFF_E2, ws + OFF_T1, ws + OFF_T2F);

  // I3: z-irfft + epilogue. waves=16384
  fno_gemm_final<<<4096, BLK, 0, stream>>>(
      ws + OFF_T2F, ws + OFF_B4, pot, W0, b0, W4, b4, out);
}